// GraphConvLayer_8933531976247
// MI455X (gfx1250) — compile-verified
//
#include <hip/hip_runtime.h>
#include <stdint.h>

// Problem constants (from reference): B=16, S=4096, F_IN=F_OUT=64
#define S_LEN   4096
#define NBATCH  16
#define F       64
#define TILE_S  128
#define HALO    20
#define XROWS   (TILE_S + 2 * HALO)   // 168 valid rows
#define XROWS_P 176                   // padded to 11 tiles of 16 for WMMA
#define LSTRIDE 68                    // 64 + 4 pad: conflict-free banks, 16B aligned

typedef __attribute__((ext_vector_type(2))) float v2f;
typedef __attribute__((ext_vector_type(4))) float v4f;
typedef __attribute__((ext_vector_type(8))) float v8f;

// Closed-form D^{-1/2}: rowsum = 1 + [s-1 ok] + [s+1 ok] + 0.5*([±5]+[±10]+[±20])
__device__ __forceinline__ float dinv_sqrt(int s) {
    float r = 1.0f;
    r += (s >= 1)        ? 1.0f : 0.0f;
    r += (s + 1 < S_LEN) ? 1.0f : 0.0f;
    r += 0.5f * (((s >= 5)  ? 1.0f : 0.0f) + ((s + 5  < S_LEN) ? 1.0f : 0.0f) +
                 ((s >= 10) ? 1.0f : 0.0f) + ((s + 10 < S_LEN) ? 1.0f : 0.0f) +
                 ((s >= 20) ? 1.0f : 0.0f) + ((s + 20 < S_LEN) ? 1.0f : 0.0f));
    return 1.0f / sqrtf(r);   // rowsum >= 1, reference's 1e-6 clamp is moot
}

// CDNA5 async global->LDS copy (ASYNCcnt tracked, no VGPR round-trip)
__device__ __forceinline__ void async_copy_b128(unsigned lds_byte_addr,
                                                const void* gaddr) {
    asm volatile("global_load_async_to_lds_b128 %0, %1, off"
                 :: "v"(lds_byte_addr), "v"(gaddr)
                 : "memory");
}
__device__ __forceinline__ void wait_asynccnt0() {
    asm volatile("s_wait_asynccnt 0x0" ::: "memory");
}

__global__ __launch_bounds__(256)
void gcn_fused_kernel(const float* __restrict__ x,
                      const float* __restrict__ W,
                      const float* __restrict__ bias,
                      float* __restrict__ out) {
    __shared__ float xs [XROWS_P * LSTRIDE];   // x tile + halo
    __shared__ float sup[XROWS_P * LSTRIDE];   // support = xW^T + b
    __shared__ float wl [F * LSTRIDE];         // W as [O][K]
    __shared__ float bl [F];
    __shared__ float dl [XROWS_P];             // D^{-1/2} per halo row (0 if OOB)

    const int tid      = threadIdx.x;
    const int batch    = blockIdx.x >> 5;          // 32 row-tiles per batch
    const int tile     = blockIdx.x & 31;
    const int rowStart = tile * TILE_S;
    const long base    = (long)batch * S_LEN * F;

    // ---- Stage x tile (+/-20 halo) into LDS via async b128 copies ----
    for (int i = tid; i < XROWS_P * (F / 4); i += 256) {
        const int r  = i / (F / 4);
        const int c4 = (i % (F / 4)) * 4;
        const int gr = rowStart - HALO + r;
        const unsigned lds = (unsigned)(uintptr_t)&xs[r * LSTRIDE + c4];
        if (r < XROWS && gr >= 0 && gr < S_LEN) {
            async_copy_b128(lds, x + base + (long)gr * F + c4);
        } else {
            *(v4f*)(&xs[r * LSTRIDE + c4]) = (v4f){0.0f, 0.0f, 0.0f, 0.0f};
        }
    }
    // ---- Stage W [O][K] via async copies; bias + dinv table directly ----
    for (int i = tid; i < F * (F / 4); i += 256) {
        const int o  = i / (F / 4);
        const int c4 = (i % (F / 4)) * 4;
        async_copy_b128((unsigned)(uintptr_t)&wl[o * LSTRIDE + c4],
                        W + o * F + c4);
    }
    if (tid < F) bl[tid] = bias[tid];
    for (int r = tid; r < XROWS_P; r += 256) {
        const int gr = rowStart - HALO + r;
        dl[r] = (gr >= 0 && gr < S_LEN) ? dinv_sqrt(gr) : 0.0f;  // 0 => masks OOB taps
    }
    wait_asynccnt0();      // own async copies landed in LDS
    __syncthreads();       // everyone's copies visible

    // ---- support = x @ W^T + b via V_WMMA_F32_16X16X4_F32 ----
    // 11 row-tiles x 4 col-tiles = 44 tiles over 8 waves (scalar loop: EXEC stays all-1s).
    const int wave = __builtin_amdgcn_readfirstlane(tid >> 5);
    const int lane = tid & 31;
    const int half = lane >> 4;    // K-pair select (ISA f32 A/B layout)
    const int l16  = lane & 15;

    for (int t = wave; t < (XROWS_P / 16) * (F / 16); t += 8) {
        const int mt = t / (F / 16);
        const int nt = t % (F / 16);
        const int m  = mt * 16 + l16;   // A-frag row
        const int n  = nt * 16 + l16;   // B-frag col == C-frag N
        v8f c = {};
#pragma unroll
        for (int kt = 0; kt < F / 4; ++kt) {
            const int k0 = kt * 4 + 2 * half;   // VGPR0: K={0,2}, VGPR1: K={1,3}
            v2f a, b;
            a.x = xs[m * LSTRIDE + k0];
            a.y = xs[m * LSTRIDE + k0 + 1];
            b.x = wl[n * LSTRIDE + k0];         // B[k][n] = W[n][k]
            b.y = wl[n * LSTRIDE + k0 + 1];
            c = __builtin_amdgcn_wmma_f32_16x16x4_f32(
                    false, a, false, b, (short)0, c, false, false);
        }
        const float bv = bl[n];
#pragma unroll
        for (int r = 0; r < 8; ++r) {           // C layout: VGPR r -> M = r + 8*half
            const int mm = mt * 16 + r + 8 * half;
            sup[mm * LSTRIDE + n] = c[r] + bv;
        }
    }
    __syncthreads();

    // ---- 9-tap banded-adjacency stencil (branch-free: OOB taps have dl==0) ----
    const int   offs[9] = {-20, -10, -5, -1, 0, 1, 5, 10, 20};
    const float aval[9] = {0.5f, 0.5f, 0.5f, 1.0f, 1.0f, 1.0f, 0.5f, 0.5f, 0.5f};

    const int fi = (tid & 15) * 4;   // float4 feature group
    const int r0 = tid >> 4;         // 16 rows per pass, 8 passes
#pragma unroll
    for (int pass = 0; pass < 8; ++pass) {
        const int i = r0 + pass * 16;       // local output row 0..127
        const float di = dl[HALO + i];
        v4f acc = {0.0f, 0.0f, 0.0f, 0.0f};
#pragma unroll
        for (int j = 0; j < 9; ++j) {
            const float cf = aval[j] * di * dl[HALO + i + offs[j]];
            const v4f v = *(const v4f*)(&sup[(HALO + i + offs[j]) * LSTRIDE + fi]);
            acc.x += cf * v.x; acc.y += cf * v.y;
            acc.z += cf * v.z; acc.w += cf * v.w;
        }
        *(v4f*)(out + base + (long)(rowStart + i) * F + fi) = acc;
    }
}

extern "C" void kernel_launch(void* const* d_in, const int* in_sizes, int n_in,
                              void* d_out, int out_size, void* d_ws, size_t ws_size,
                              hipStream_t stream) {
    const float* x = (const float*)d_in[0];   // [16, 4096, 64] f32
    const float* W = (const float*)d_in[1];   // [64, 64] f32
    const float* b = (const float*)d_in[2];   // [64] f32
    float* out = (float*)d_out;               // [16, 4096, 64] f32

    dim3 grid(NBATCH * (S_LEN / TILE_S));     // 512 workgroups
    gcn_fused_kernel<<<grid, 256, 0, stream>>>(x, W, b, out);
}